// Dis_model_60129542275
// MI455X (gfx1250) — compile-verified
//
#include <hip/hip_runtime.h>

// Optical-flow bilinear warp for MI455X (gfx1250).
// Memory-bound gather: ~503 MB of traffic, ~22 us floor @ 23.3 TB/s.
// Strategy:
//  - 3D launch (x-tile, y, b): zero index-decompose ALU, no tail branch.
//  - 2 pixels per thread (contiguous half-rows): 8 outstanding B128 gathers,
//    all flow loads / out stores stay unit-stride per wave.
//  - NT hints on streaming flow loads + out stores so the 201 MB source array
//    keeps residency priority in the 192 MB L2.
//  - global_prefetch_b8 on the flow stream.

typedef float v2f __attribute__((ext_vector_type(2)));
typedef float v4f __attribute__((ext_vector_type(4)));

__device__ __forceinline__ v4f warp_one(const v4f* __restrict__ sb,
                                        int x, int y, v2f f,
                                        int H, int W) {
    const float wy = (float)y + f.x;   // flow[...,0] = dy
    const float wx = (float)x + f.y;   // flow[...,1] = dx

    const float fy = floorf(wy);
    const float fx = floorf(wx);
    const float w_right = wx - fx;
    const float w_down  = wy - fy;
    const float w_left  = 1.0f - w_right;
    const float w_up    = 1.0f - w_down;

    int x0 = (int)fx;
    int y0 = (int)fy;
    int x1 = (int)ceilf(wx);
    int y1 = (int)ceilf(wy);

    // Clamp only; if the warp center is in-range all taps are in-range,
    // otherwise the final mask zeroes the result (matches safe-gather + mask).
    x0 = min(max(x0, 0), W - 1);
    x1 = min(max(x1, 0), W - 1);
    y0 = min(max(y0, 0), H - 1);
    y1 = min(max(y1, 0), H - 1);

    // Four 16-byte taps -> four global_load_b128 (regular temporal: we want
    // these resident in the 192 MB L2).
    const v4f g00 = sb[y0 * W + x0];
    const v4f g01 = sb[y0 * W + x1];
    const v4f g10 = sb[y1 * W + x0];
    const v4f g11 = sb[y1 * W + x1];

    const float m = (wy >= 0.0f && wy <= (float)(H - 1) &&
                     wx >= 0.0f && wx <= (float)(W - 1)) ? 1.0f : 0.0f;

    const float a00 = w_left  * w_up   * m;
    const float a01 = w_right * w_up   * m;
    const float a10 = w_left  * w_down * m;
    const float a11 = w_right * w_down * m;

    return g00 * a00 + g01 * a01 + g10 * a10 + g11 * a11;
}

__global__ __launch_bounds__(256) void warp_bilinear_kernel(
    const float* __restrict__ src,   // [B,H,W,4]
    const float* __restrict__ flow,  // [B,H,W,2]
    float* __restrict__ out)         // [B,H,W,4]
{
    constexpr int H = 768;
    constexpr int W = 1024;

    const int tid = threadIdx.x;
    const int y   = blockIdx.y;
    const int b   = blockIdx.z;

    // Each block covers 512 consecutive x as two contiguous half-rows of 256.
    const int xA = blockIdx.x * 512 + tid;
    const int xB = xA + 256;

    const int rowBase = (b * H + y) * W;
    const int idxA = rowBase + xA;
    const int idxB = rowBase + xB;

    // Prefetch the flow stream ~32 KB ahead (speculative, GL2).
    __builtin_prefetch(flow + 2 * idxA + 8192, 0, 0);

    // Streaming read-once flow: non-temporal b64 loads.
    const v2f fA = __builtin_nontemporal_load((const v2f*)flow + idxA);
    const v2f fB = __builtin_nontemporal_load((const v2f*)flow + idxB);

    const v4f* sb = (const v4f*)src + (size_t)b * (size_t)(H * W);

    const v4f rA = warp_one(sb, xA, y, fA, H, W);
    const v4f rB = warp_one(sb, xB, y, fB, H, W);

    // Streaming write-once output: non-temporal B128 stores, unit stride.
    __builtin_nontemporal_store(rA, (v4f*)out + idxA);
    __builtin_nontemporal_store(rB, (v4f*)out + idxB);
}

extern "C" void kernel_launch(void* const* d_in, const int* in_sizes, int n_in,
                              void* d_out, int out_size, void* d_ws, size_t ws_size,
                              hipStream_t stream) {
    const float* src  = (const float*)d_in[0];   // [16,768,1024,4] fp32
    const float* flow = (const float*)d_in[1];   // [16,768,1024,2] fp32
    float* out = (float*)d_out;                  // [16,768,1024,4] fp32

    constexpr int B = 16, H = 768, W = 1024;
    (void)in_sizes; (void)n_in; (void)out_size; (void)d_ws; (void)ws_size;

    dim3 grid(W / 512, H, B);    // (2, 768, 16) = 24576 blocks, no tail
    dim3 block(256);             // 8 wave32 waves

    warp_bilinear_kernel<<<grid, block, 0, stream>>>(src, flow, out);
}